// Decoder_65000035058347
// MI455X (gfx1250) — compile-verified
//
#include <hip/hip_runtime.h>
#include <hip/hip_bf16.h>

// ---------------------------------------------------------------------------
// Tacotron2-style decoder for MI455X (gfx1250, wave32).
// GEMMs use V_WMMA_F32_16X16X4_F32 (fp32 in/out, exact vs fp32 reference).
// LSTM gate GEMMs stage the shared 16xK activation matrix into LDS with the
// CDNA5 async global->LDS path (ASYNCcnt + double buffering) and read WMMA
// A-operands from LDS while streaming weights (B) from L2.
// ---------------------------------------------------------------------------

typedef float v2f __attribute__((ext_vector_type(2)));
typedef float v8f __attribute__((ext_vector_type(8)));
typedef int   v4i __attribute__((vector_size(16)));
typedef __attribute__((address_space(1))) v4i* gptr_v4i;
typedef __attribute__((address_space(3))) v4i* lptr_v4i;

#define NB       16      // batch
#define TENC     512
#define DENC     512
#define NMEL     80
#define NFPS     3
#define FRAME    240     // NMEL*NFPS
#define NSTEPS   200     // F_MEL / NFPS
#define PRE      256
#define HRNN     1024
#define NGATE    4096
#define ADIM     128
#define LFILT    32
#define LK       31
#define KCHUNK   256     // K-chunk staged in LDS (divides all segment bounds)

#if defined(__has_builtin)
#if __has_builtin(__builtin_amdgcn_global_load_async_to_lds_b128) && \
    __has_builtin(__builtin_amdgcn_s_wait_asynccnt)
#define HAVE_ASYNC_LDS 1
#endif
#endif
#ifndef HAVE_ASYNC_LDS
#define HAVE_ASYNC_LDS 0
#endif

// Copy 16B global -> LDS. Async (ASYNCcnt-tracked) when available.
__device__ __forceinline__ void copy_f4_to_lds(const float* __restrict__ g,
                                               float* __restrict__ l) {
#if HAVE_ASYNC_LDS
  __builtin_amdgcn_global_load_async_to_lds_b128(
      (gptr_v4i)(unsigned long long)g,
      (lptr_v4i)(unsigned)(unsigned long long)l,
      0, 0);
#else
  *(float4*)l = *(const float4*)g;
#endif
}

#if HAVE_ASYNC_LDS
#define WAIT_ASYNC(n) __builtin_amdgcn_s_wait_asynccnt(n)
#else
#define WAIT_ASYNC(n)
#endif

// fp32 WMMA 16x16x4 inner loop over a K-span.
// A (16x4): lanes 0-15 hold M rows, VGPR pair = K {kb,kb+1}, lane-half -> kb.
// B (4x16): lanes hold N columns, same K split.
__device__ __forceinline__ v8f wmma_span(v8f acc, const float* __restrict__ ap,
                                         const float* __restrict__ bp, int K) {
  for (int k = 0; k < K; k += 4) {
    v2f a = *(const v2f*)(ap + k);
    v2f b = *(const v2f*)(bp + k);
    acc = __builtin_amdgcn_wmma_f32_16x16x4_f32(false, a, false, b,
                                                (short)0, acc, false, false);
  }
  return acc;
}

// ---------------------------------------------------------------------------
// Generic GEMM: C[M,N] = act(A[M,K] @ W[N,K]^T + bias), one wave per 16x16 tile
// ---------------------------------------------------------------------------
__global__ void gemm_kernel(const float* __restrict__ A, int lda,
                            const float* __restrict__ W, int ldw,
                            const float* __restrict__ bias,
                            float* __restrict__ C, int ldc,
                            int n_tiles, int K, int relu) {
  int tile = blockIdx.x;
  int mt = tile / n_tiles, nt = tile % n_tiles;
  int lane = threadIdx.x;
  int mrow = lane & 15;
  int kb   = (lane >> 4) * 2;
  int n    = nt * 16 + (lane & 15);
  const float* ap = A + (long)(mt * 16 + mrow) * lda + kb;
  const float* bp = W + (long)n * ldw + kb;
  v8f acc = {};
  acc = wmma_span(acc, ap, bp, K);
  int mhalf = (lane >> 4) * 8;
  float bv = bias ? bias[n] : 0.0f;
  #pragma unroll
  for (int r = 0; r < 8; ++r) {
    float v = acc[r] + bv;
    if (relu) v = fmaxf(v, 0.0f);
    C[(long)(mt * 16 + mhalf + r) * ldc + n] = v;
  }
}

// ---------------------------------------------------------------------------
// LSTM gate GEMM: gates(16,4096) = [a0|a1|a2] @ [w_ih | w_hh]^T + b_ih + b_hh
// a0,a1 feed w_ih (cols [0,K0+K1)), a2 feeds w_hh.
// 8 waves/block, 8 N-tiles/block; A staged per K-chunk in LDS (double buffer,
// async global->LDS); B streamed from global. K0,K1,K2 multiples of KCHUNK.
// ---------------------------------------------------------------------------
__global__ void gates_kernel(const float* __restrict__ a0, int lda0, int K0,
                             const float* __restrict__ a1, int lda1, int K1,
                             const float* __restrict__ a2, int lda2, int K2,
                             const float* __restrict__ w_ih, int ld_ih,
                             const float* __restrict__ w_hh, int ld_hh,
                             const float* __restrict__ b_ih,
                             const float* __restrict__ b_hh,
                             float* __restrict__ gates) {
  __shared__ float sA[2][16 * KCHUNK];   // 2 x 16 KB
  const int tid  = threadIdx.x;          // 0..255 (8 waves)
  const int lane = tid & 31;
  const int wave = tid >> 5;
  const int nt   = blockIdx.x * 8 + wave;
  const int n    = nt * 16 + (lane & 15);
  const int mrow = lane & 15;
  const int kb   = (lane >> 4) * 2;
  const int Ktot = K0 + K1 + K2;
  const int C    = Ktot / KCHUNK;

  // Stage one 16xKCHUNK chunk of the segmented A into LDS (16B per thread x4).
  auto stage = [&](float* dst, int kg) {
    const float* seg; int ld, ks;
    if (kg < K0)           { seg = a0; ld = lda0; ks = kg; }
    else if (kg < K0 + K1) { seg = a1; ld = lda1; ks = kg - K0; }
    else                   { seg = a2; ld = lda2; ks = kg - K0 - K1; }
    #pragma unroll
    for (int i = 0; i < 4; ++i) {
      int off = (tid + 256 * i) * 4;     // float offset in chunk
      int m = off >> 8;                  // / KCHUNK
      int k = off & (KCHUNK - 1);
      copy_f4_to_lds(seg + (long)m * ld + ks + k, dst + off);
    }
  };

  v8f acc = {};
  stage(&sA[0][0], 0);
  for (int c = 0; c < C; ++c) {
    if (c + 1 < C) {
      stage(&sA[(c + 1) & 1][0], (c + 1) * KCHUNK);
      WAIT_ASYNC(4);                     // chunk c complete (in-order returns)
    } else {
      WAIT_ASYNC(0);
    }
    __syncthreads();                     // all waves' pieces visible

    int kg = c * KCHUNK;
    const float* wmat; int ldwm, wcol;
    if (kg < K0 + K1) { wmat = w_ih; ldwm = ld_ih; wcol = kg; }
    else              { wmat = w_hh; ldwm = ld_hh; wcol = kg - K0 - K1; }

    const float* ap = &sA[c & 1][mrow * KCHUNK + kb];       // ds_load_b64
    const float* bp = wmat + (long)n * ldwm + wcol + kb;    // global b64
    acc = wmma_span(acc, ap, bp, KCHUNK);
    __syncthreads();                     // done reading before buf reuse
  }

  int mhalf = (lane >> 4) * 8;
  float bv = b_ih[n] + b_hh[n];
  #pragma unroll
  for (int r = 0; r < 8; ++r)
    gates[(long)(mhalf + r) * NGATE + n] = acc[r] + bv;
}

// ---------------------------------------------------------------------------
// LSTM pointwise: i,f,g,o split; updates h,c in place.
// ---------------------------------------------------------------------------
__global__ void lstm_kernel(const float* __restrict__ gates,
                            float* __restrict__ h, float* __restrict__ c) {
  int idx = blockIdx.x * 256 + threadIdx.x;
  if (idx >= NB * HRNN) return;
  int b = idx >> 10, n = idx & (HRNN - 1);
  const float* g = gates + (long)b * NGATE;
  float gi = g[n], gf = g[HRNN + n], gg = g[2 * HRNN + n], go = g[3 * HRNN + n];
  float si = 1.0f / (1.0f + expf(-gi));
  float sf = 1.0f / (1.0f + expf(-gf));
  float so = 1.0f / (1.0f + expf(-go));
  float cn = sf * c[idx] + si * tanhf(gg);
  c[idx] = cn;
  h[idx] = so * tanhf(cn);
}

// ---------------------------------------------------------------------------
// Location-sensitive attention: one block per batch element.
// conv(w, w_cum) -> loc; e = tanh(q + loc@dense^T + pm) @ v; softmax; ctx.
// ---------------------------------------------------------------------------
__global__ void attention_kernel(const float* __restrict__ h_a,
                                 const float* __restrict__ Wq,
                                 const float* __restrict__ v_att,
                                 const float* __restrict__ conv_w,   // (32,2,31)
                                 const float* __restrict__ dense_w,  // (128,32)
                                 const float* __restrict__ pm,       // (16*512,128)
                                 const float* __restrict__ memory,   // (16*512,512)
                                 const int*   __restrict__ mlen,
                                 float* __restrict__ w_st,
                                 float* __restrict__ wcum_st,
                                 float* __restrict__ ctx_st,
                                 float* __restrict__ align_out,
                                 int s) {
  __shared__ float q[ADIM];
  __shared__ float sw[TENC], swc[TENC], se[TENC];
  __shared__ float red[256];
  int b = blockIdx.x, tid = threadIdx.x;

  for (int t = tid; t < TENC; t += 256) {
    sw[t]  = w_st[b * TENC + t];
    swc[t] = wcum_st[b * TENC + t];
  }
  if (tid < ADIM) {
    const float* hr = h_a + (long)b * HRNN;
    const float* wr = Wq + (long)tid * HRNN;
    float acc = 0.0f;
    for (int k = 0; k < HRNN; ++k) acc += hr[k] * wr[k];
    q[tid] = acc;
  }
  __syncthreads();

  int L = mlen[b];
  for (int t = tid; t < TENC; t += 256) {
    float e;
    if (t >= L) {
      e = -1.0e9f;
    } else {
      float loc[LFILT];
      #pragma unroll 4
      for (int c = 0; c < LFILT; ++c) {
        const float* w0 = conv_w + c * (2 * LK);
        const float* w1 = w0 + LK;
        float a = 0.0f;
        for (int kk = 0; kk < LK; ++kk) {
          int tt = t + kk - (LK - 1) / 2;
          if (tt >= 0 && tt < TENC) a += sw[tt] * w0[kk] + swc[tt] * w1[kk];
        }
        loc[c] = a;
      }
      const float* pmr = pm + (long)(b * TENC + t) * ADIM;
      float acc = 0.0f;
      for (int f = 0; f < ADIM; ++f) {
        const float* dw = dense_w + f * LFILT;
        float pl = 0.0f;
        #pragma unroll
        for (int c = 0; c < LFILT; ++c) pl += loc[c] * dw[c];
        acc += v_att[f] * tanhf(q[f] + pl + pmr[f]);
      }
      e = acc;
    }
    se[t] = e;
  }
  __syncthreads();

  // softmax over T=512 (each thread owns t and t+256)
  float lm = fmaxf(se[tid], se[tid + 256]);
  red[tid] = lm;
  __syncthreads();
  for (int off = 128; off > 0; off >>= 1) {
    if (tid < off) red[tid] = fmaxf(red[tid], red[tid + off]);
    __syncthreads();
  }
  float mx = red[0];
  __syncthreads();
  float ex0 = expf(se[tid] - mx), ex1 = expf(se[tid + 256] - mx);
  red[tid] = ex0 + ex1;
  __syncthreads();
  for (int off = 128; off > 0; off >>= 1) {
    if (tid < off) red[tid] += red[tid + off];
    __syncthreads();
  }
  float inv = 1.0f / red[0];
  __syncthreads();
  se[tid] = ex0 * inv;
  se[tid + 256] = ex1 * inv;
  __syncthreads();

  for (int t = tid; t < TENC; t += 256) {
    float wn = se[t];
    w_st[b * TENC + t]    = wn;
    wcum_st[b * TENC + t] = swc[t] + wn;
    align_out[(long)b * (NSTEPS * TENC) + (long)s * TENC + t] = wn;
  }
  __syncthreads();

  // ctx[d] = sum_t w_new[t] * memory[b,t,d]
  for (int d = tid; d < DENC; d += 256) {
    const float* mb = memory + (long)b * TENC * DENC + d;
    float acc = 0.0f;
    for (int t = 0; t < TENC; ++t) acc += se[t] * mb[(long)t * DENC];
    ctx_st[b * DENC + d] = acc;
  }
}

// ---------------------------------------------------------------------------
// Projection: out(16,240) = [h_d | ctx] @ proj_w^T + b, scattered into the
// (B, 80, 600) mel output layout. One wave per 16-wide N tile (15 tiles).
// ---------------------------------------------------------------------------
__global__ void proj_kernel(const float* __restrict__ h_d,
                            const float* __restrict__ ctx,
                            const float* __restrict__ proj_w,  // (240,1536)
                            const float* __restrict__ proj_b,
                            float* __restrict__ mel_out, int s) {
  int nt = blockIdx.x;
  int lane = threadIdx.x;
  int mrow = lane & 15;
  int kb   = (lane >> 4) * 2;
  int n    = nt * 16 + (lane & 15);
  v8f acc = {};
  acc = wmma_span(acc, h_d + (long)mrow * HRNN + kb,
                  proj_w + (long)n * (HRNN + DENC) + kb, HRNN);
  acc = wmma_span(acc, ctx + (long)mrow * DENC + kb,
                  proj_w + (long)n * (HRNN + DENC) + HRNN + kb, DENC);
  int mhalf = (lane >> 4) * 8;
  float bv = proj_b[n];
  int m = n % NMEL, fps = n / NMEL;
  #pragma unroll
  for (int r = 0; r < 8; ++r) {
    int b = mhalf + r;
    mel_out[(long)b * (NMEL * NSTEPS * NFPS) + (long)m * (NSTEPS * NFPS)
            + s * NFPS + fps] = acc[r] + bv;
  }
}

// ---------------------------------------------------------------------------
// Build prenet input frames: x[s,b,j]; s=0 is the <go> zero frame,
// s>=1 picks decoder_inputs[b, j%80, (s-1)*3 + j/80].
// ---------------------------------------------------------------------------
__global__ void build_x_kernel(const float* __restrict__ dec_in,
                               float* __restrict__ xcat) {
  int idx = blockIdx.x * 256 + threadIdx.x;
  if (idx >= NSTEPS * NB * FRAME) return;
  int j = idx % FRAME;
  int b = (idx / FRAME) % NB;
  int s = idx / (FRAME * NB);
  float val = 0.0f;
  if (s > 0) {
    int m = j % NMEL, fps = j / NMEL;
    val = dec_in[(long)b * NMEL * (NSTEPS * NFPS) + (long)m * (NSTEPS * NFPS)
                 + (s - 1) * NFPS + fps];
  }
  xcat[idx] = val;
}

__global__ void zero_kernel(float* __restrict__ p, int n) {
  int i = blockIdx.x * 256 + threadIdx.x;
  if (i < n) p[i] = 0.0f;
}

// ---------------------------------------------------------------------------
extern "C" void kernel_launch(void* const* d_in, const int* in_sizes, int n_in,
                              void* d_out, int out_size, void* d_ws, size_t ws_size,
                              hipStream_t stream) {
  (void)in_sizes; (void)n_in; (void)out_size; (void)ws_size;

  const float* memory      = (const float*)d_in[0];
  const float* dec_inputs  = (const float*)d_in[1];
  const int*   mem_len     = (const int*)  d_in[2];
  const float* prenet_w1   = (const float*)d_in[3];
  const float* prenet_w2   = (const float*)d_in[4];
  const float* att_w_ih    = (const float*)d_in[5];
  const float* att_w_hh    = (const float*)d_in[6];
  const float* att_b_ih    = (const float*)d_in[7];
  const float* att_b_hh    = (const float*)d_in[8];
  const float* dec_w_ih    = (const float*)d_in[9];
  const float* dec_w_hh    = (const float*)d_in[10];
  const float* dec_b_ih    = (const float*)d_in[11];
  const float* dec_b_hh    = (const float*)d_in[12];
  const float* Wq          = (const float*)d_in[13];
  const float* Wm          = (const float*)d_in[14];
  const float* v_att       = (const float*)d_in[15];
  const float* loc_conv_w  = (const float*)d_in[16];
  const float* loc_dense_w = (const float*)d_in[17];
  const float* proj_w      = (const float*)d_in[18];
  const float* proj_b      = (const float*)d_in[19];

  float* mel_out   = (float*)d_out;                       // (16, 80, 600)
  float* align_out = mel_out + NB * NMEL * NSTEPS * NFPS; // (16, 200, 512)

  // Workspace layout (floats)
  float* ws    = (float*)d_ws;
  float* pm    = ws;                                   // 16*512*128 = 1048576
  float* xcat  = pm + NB * TENC * ADIM;                // 200*16*240 =  768000
  float* pre1  = xcat + NSTEPS * NB * FRAME;           // 200*16*256 =  819200
  float* pre2  = pre1 + NSTEPS * NB * PRE;             //               819200
  float* h_a   = pre2 + NSTEPS * NB * PRE;             // 16*1024
  float* c_a   = h_a + NB * HRNN;
  float* h_d   = c_a + NB * HRNN;
  float* c_d   = h_d + NB * HRNN;
  float* w_st  = c_d + NB * HRNN;                      // 16*512
  float* wc_st = w_st + NB * TENC;
  float* ctx   = wc_st + NB * TENC;
  float* gates = ctx + NB * DENC;                      // 16*4096

  // Zero recurrent state (h_a..ctx contiguous)
  int state_n = 4 * NB * HRNN + 3 * NB * TENC;
  zero_kernel<<<(state_n + 255) / 256, 256, 0, stream>>>(h_a, state_n);

  // Prenet inputs for all steps, then two relu GEMMs (M=3200)
  int xn = NSTEPS * NB * FRAME;
  build_x_kernel<<<(xn + 255) / 256, 256, 0, stream>>>(dec_inputs, xcat);
  gemm_kernel<<<NSTEPS * (PRE / 16), 32, 0, stream>>>(
      xcat, FRAME, prenet_w1, FRAME, nullptr, pre1, PRE, PRE / 16, FRAME, 1);
  gemm_kernel<<<NSTEPS * (PRE / 16), 32, 0, stream>>>(
      pre1, PRE, prenet_w2, PRE, nullptr, pre2, PRE, PRE / 16, PRE, 1);

  // processed_memory = memory @ Wm^T : (8192,512) x (128,512)^T
  gemm_kernel<<<(NB * TENC / 16) * (ADIM / 16), 32, 0, stream>>>(
      memory, DENC, Wm, DENC, nullptr, pm, ADIM, ADIM / 16, DENC, 0);

  // Sequential decode
  for (int s = 0; s < NSTEPS; ++s) {
    // Attention LSTM gates: A = [pre_s(256) | ctx(512) | h_a(1024)]
    gates_kernel<<<NGATE / 128, 256, 0, stream>>>(
        pre2 + (long)s * NB * PRE, PRE, PRE,
        ctx, DENC, DENC,
        h_a, HRNN, HRNN,
        att_w_ih, PRE + DENC, att_w_hh, HRNN,
        att_b_ih, att_b_hh, gates);
    lstm_kernel<<<(NB * HRNN + 255) / 256, 256, 0, stream>>>(gates, h_a, c_a);

    attention_kernel<<<NB, 256, 0, stream>>>(
        h_a, Wq, v_att, loc_conv_w, loc_dense_w, pm, memory, mem_len,
        w_st, wc_st, ctx, align_out, s);

    // Decoder LSTM gates: A = [h_a(1024) | ctx(512) | h_d(1024)]
    gates_kernel<<<NGATE / 128, 256, 0, stream>>>(
        h_a, HRNN, HRNN,
        ctx, DENC, DENC,
        h_d, HRNN, HRNN,
        dec_w_ih, HRNN + DENC, dec_w_hh, HRNN,
        dec_b_ih, dec_b_hh, gates);
    lstm_kernel<<<(NB * HRNN + 255) / 256, 256, 0, stream>>>(gates, h_d, c_d);

    proj_kernel<<<FRAME / 16, 32, 0, stream>>>(h_d, ctx, proj_w, proj_b,
                                               mel_out, s);
  }
}